// DetectionCriterion_86517821214147
// MI455X (gfx1250) — compile-verified
//
#include <hip/hip_runtime.h>
#include <hip/hip_bf16.h>
#include <stdint.h>

// ---------------- problem geometry ----------------
#define N_TPL     25
#define BATCH     32
#define PIX       16384                   // 128*128
#define PB_CLASS  (N_TPL * PIX)           // 409600 class elems / batch
#define SB_OUT    (5 * N_TPL * PIX)       // 2048000 output elems / batch
#define PB_REG    (4 * N_TPL * PIX)       // 1638400 reg-map elems / batch
#define KMAX      128                     // pos_max == neg_max == 128
#define HNEG      0.03f

// ---------------- rank-select config ----------------
#define NBINS_T   8192                    // 13-bit priority bins
#define SLICE     128                     // only histogram top 1/64 of bins
#define SLICE_MIN (NBINS_T - SLICE)       // 8064

// ---------------- launch geometry ----------------
#define TPB       256
#define VEC       4
#define EPI       (TPB * VEC)             // 1024 elems per pipeline stage
#define ITERS     8
#define BLK_ELEMS (EPI * ITERS)           // 8192 elems per block
#define BLOCKS_X  (PB_CLASS / BLK_ELEMS)  // 50

__device__ __forceinline__ float softplusf(float x) {
    // numerically stable log1p(exp(x))
    return fmaxf(x, 0.f) + log1pf(expf(-fabsf(x)));
}

__device__ __forceinline__ unsigned prio_hash(unsigned b, unsigned i) {
    unsigned x = i + b * 0x9E3779B9u + 0x85EBCA6Bu;
    x ^= x >> 16; x *= 0x7FEB352Du;
    x ^= x >> 15; x *= 0x846CA68Bu;
    x ^= x >> 16;
    return x;
}

// =====================================================================
// Pass 1: hard-negative mining + top-slice priority histogram.
// Streams class_map + classification through a double-buffered
// global_load_async_to_lds_b128 pipeline (lane-private LDS slots).
// =====================================================================
__global__ __launch_bounds__(TPB) void k_mine_hist(
    const float* __restrict__ out_buf, const float* __restrict__ cls_map,
    unsigned* __restrict__ hist)
{
    const int      b        = blockIdx.y;
    const int      tid      = threadIdx.x;
    const unsigned blk_base = (unsigned)blockIdx.x * BLK_ELEMS;

    const uint64_t labp = (uint64_t)(uintptr_t)(cls_map + (size_t)b * PB_CLASS + blk_base);
    const uint64_t clsp = (uint64_t)(uintptr_t)(out_buf + (size_t)b * SB_OUT   + blk_base);

    __shared__ float s_lab[2][EPI];
    __shared__ float s_cls[2][EPI];

    const unsigned lane_byte = (unsigned)tid * (VEC * 4u);
    unsigned lds_lab[2], lds_cls[2];
    lds_lab[0] = (unsigned)(size_t)&s_lab[0][tid * VEC];
    lds_lab[1] = (unsigned)(size_t)&s_lab[1][tid * VEC];
    lds_cls[0] = (unsigned)(size_t)&s_cls[0][tid * VEC];
    lds_cls[1] = (unsigned)(size_t)&s_cls[1][tid * VEC];

    auto stage = [&](int it, int buf) {
        unsigned voff = (unsigned)it * (EPI * 4u) + lane_byte;
        asm volatile(
            "global_load_async_to_lds_b128 %0, %2, %3\n\t"
            "global_load_async_to_lds_b128 %1, %2, %4"
            :
            : "v"(lds_lab[buf]), "v"(lds_cls[buf]), "v"(voff), "s"(labp), "s"(clsp)
            : "memory");
    };

    stage(0, 0);
    for (int it = 0; it < ITERS; ++it) {
        const int buf = it & 1;
        if (it + 1 < ITERS) {
            stage(it + 1, buf ^ 1);
            asm volatile("s_wait_asynccnt 2" ::: "memory");  // oldest stage done
        } else {
            asm volatile("s_wait_asynccnt 0" ::: "memory");
        }
        float4 lab4 = *(const float4*)&s_lab[buf][tid * VEC];
        float4 cls4 = *(const float4*)&s_cls[buf][tid * VEC];
        float labv[4] = {lab4.x, lab4.y, lab4.z, lab4.w};
        float clsv[4] = {cls4.x, cls4.y, cls4.z, cls4.w};
        const unsigned ibase = blk_base + (unsigned)it * EPI + (unsigned)tid * VEC;
#pragma unroll
        for (int j = 0; j < 4; ++j) {
            float lb = labv[j], c = clsv[j];
            if (lb == 0.f) continue;
            float sp = softplusf(-lb * c);
            if (sp < HNEG) continue;                      // mined away
            unsigned bin = prio_hash((unsigned)b, ibase + (unsigned)j) >> 19;
            if (bin >= SLICE_MIN) {                       // only top 1/64 matters for rank-128
                int s = (lb > 0.f) ? 0 : 1;
                atomicAdd(&hist[(unsigned)(b * 2 + s) * SLICE + (bin - SLICE_MIN)], 1u);
            }
        }
    }
}

// =====================================================================
// Pass 2: per (batch, sign) scan top slice for the rank-128 threshold
// bin T and residual slot count R. One thread per (b,sign).
// =====================================================================
__global__ void k_thresh(const unsigned* __restrict__ hist,
                         int* __restrict__ thresh, int* __restrict__ slots)
{
    int i = threadIdx.x;
    if (i >= BATCH * 2) return;
    const unsigned* h = hist + (unsigned)i * SLICE;
    int run = 0, T = -1, R = 0;                           // T==-1 -> keep everything
    for (int bin = SLICE - 1; bin >= 0; --bin) {
        int c = (int)h[bin];
        if (run + c >= KMAX) { T = bin + SLICE_MIN; R = KMAX - run; break; }
        run += c;
    }
    thresh[i] = T;
    slots[i]  = R;
}

// =====================================================================
// Pass 3: final loss. Re-reads the (L2-resident) class pair, applies the
// keep rule, accumulates softplus class loss + sparse smooth-L1 gathers,
// wave32 shuffle -> LDS -> global_atomic_add_f32.
// =====================================================================
__global__ __launch_bounds__(TPB) void k_loss(
    const float* __restrict__ out_buf, const float* __restrict__ cls_map,
    const float* __restrict__ reg_map, const int* __restrict__ thresh,
    int* __restrict__ slots, float* __restrict__ out)
{
    const int      b        = blockIdx.y;
    const int      tid      = threadIdx.x;
    const unsigned blk_base = (unsigned)blockIdx.x * BLK_ELEMS;

    const int Tpos = thresh[b * 2 + 0];
    const int Tneg = thresh[b * 2 + 1];
    int* slot_pos = slots + b * 2 + 0;
    int* slot_neg = slots + b * 2 + 1;

    const float* labB = cls_map + (size_t)b * PB_CLASS;
    const float* clsB = out_buf + (size_t)b * SB_OUT;
    float acc = 0.f;

    for (int it = 0; it < ITERS; ++it) {
        const unsigned ibase = blk_base + (unsigned)it * EPI + (unsigned)tid * VEC;
        float4 lab4 = *(const float4*)(labB + ibase);
        float4 cls4 = *(const float4*)(clsB + ibase);
        float labv[4] = {lab4.x, lab4.y, lab4.z, lab4.w};
        float clsv[4] = {cls4.x, cls4.y, cls4.z, cls4.w};
#pragma unroll
        for (int j = 0; j < 4; ++j) {
            float lb = labv[j], c = clsv[j];
            if (lb == 0.f) continue;
            float sp = softplusf(-lb * c);                // == class loss if kept
            if (sp < HNEG) continue;
            const unsigned i   = ibase + (unsigned)j;
            const int      bin = (int)(prio_hash((unsigned)b, i) >> 19);
            const bool     pos = lb > 0.f;
            const int      T   = pos ? Tpos : Tneg;
            bool keep;
            if (bin > T)        keep = true;              // also covers T==-1 keep-all
            else if (bin == T)  keep = atomicSub(pos ? slot_pos : slot_neg, 1) > 0;
            else                keep = false;
            if (!keep) continue;
            acc += sp;
            if (pos) {                                    // sparse smooth-L1 gather
                const unsigned t   = i / PIX;
                const unsigned pix = i % PIX;
                const float* rp = clsB + (size_t)(N_TPL + t) * PIX + pix;      // output[:,25+r*25+t]
                const float* mp = reg_map + (size_t)b * PB_REG + (size_t)t * PIX + pix;
#pragma unroll
                for (int r = 0; r < 4; ++r) {
                    float d  = rp[(size_t)r * N_TPL * PIX] - mp[(size_t)r * N_TPL * PIX];
                    float ad = fabsf(d);
                    acc += (ad < 1.f) ? 0.5f * d * d : (ad - 0.5f);
                }
            }
        }
    }

    // wave32 reduction -> LDS -> one f32 atomic per block
    for (int off = 16; off; off >>= 1) acc += __shfl_down(acc, off, 32);
    __shared__ float wsum[TPB / 32];
    const int lane = tid & 31, w = tid >> 5;
    if (lane == 0) wsum[w] = acc;
    __syncthreads();
    if (w == 0) {
        acc = (lane < (TPB / 32)) ? wsum[lane] : 0.f;
        for (int off = 4; off; off >>= 1) acc += __shfl_down(acc, off, 32);
        if (lane == 0) atomicAdd(out, acc);
    }
}

extern "C" void kernel_launch(void* const* d_in, const int* in_sizes, int n_in,
                              void* d_out, int out_size, void* d_ws, size_t ws_size,
                              hipStream_t stream)
{
    (void)in_sizes; (void)n_in; (void)out_size; (void)ws_size;
    const float* out_buf = (const float*)d_in[0];   // output  (B,125,H,W)
    const float* cls_map = (const float*)d_in[1];   // class_map (B,25,H,W)
    const float* reg_map = (const float*)d_in[2];   // regression_map (B,100,H,W)
    float* out = (float*)d_out;

    unsigned* hist  = (unsigned*)d_ws;                               // 64*128 u32 = 32 KB
    int*      thrs  = (int*)((char*)d_ws + (size_t)BATCH * 2 * SLICE * 4);
    int*      slots = thrs + BATCH * 2;

    hipMemsetAsync(d_ws, 0, (size_t)BATCH * 2 * SLICE * 4, stream);
    hipMemsetAsync(d_out, 0, sizeof(float), stream);

    dim3 grid(BLOCKS_X, BATCH);
    k_mine_hist<<<grid, TPB, 0, stream>>>(out_buf, cls_map, hist);
    k_thresh<<<1, 64, 0, stream>>>(hist, thrs, slots);
    k_loss<<<grid, TPB, 0, stream>>>(out_buf, cls_map, reg_map, thrs, slots, out);
}